// ARD_58256936402966
// MI455X (gfx1250) — compile-verified
//
#include <hip/hip_runtime.h>
#include <hip/hip_bf16.h>

typedef float v2f __attribute__((ext_vector_type(2)));
typedef float v8f __attribute__((ext_vector_type(8)));

#define BM 128
#define BN 128
#define DD 64
#define LDX 68   // padded LDS row stride (dwords): conflict-free WMMA fragment loads, keeps 16B alignment

__global__ __launch_bounds__(256) void ARD_wmma_f32_kernel(
    const float* __restrict__ x,    // [8192][64]
    const float* __restrict__ y,    // [8192][64]
    const float* __restrict__ lbw,  // [64]
    float* __restrict__ out)        // [8192][8192]
{
    __shared__ float sX[BM * LDX];   // weighted x rows (wx = x * bw)
    __shared__ float sY[BN * LDX];   // raw y rows
    __shared__ float sBW[DD];
    __shared__ float sX2[BM];
    __shared__ float sY2[BN];
    __shared__ float sPartX[256];
    __shared__ float sPartY[256];

    const int tid = threadIdx.x;
    const int blockRow = blockIdx.y * BM;   // x rows
    const int blockCol = blockIdx.x * BN;   // y rows

    // ---- stage bandwidths ----
    if (tid < DD) sBW[tid] = __expf(lbw[tid]);
    __syncthreads();

    // ---- stage x (weighted) + y (raw), accumulate partial norms ----
    {
        const int r = tid >> 1;         // 0..127
        const int h = tid & 1;          // half-row
        const int dbase = h * 32;

        const float4* xrow = (const float4*)(x + (size_t)(blockRow + r) * DD + dbase);
        const float4* yrow = (const float4*)(y + (size_t)(blockCol + r) * DD + dbase);

        float px = 0.f, py = 0.f;
        #pragma unroll
        for (int i = 0; i < 8; ++i) {
            const int d = dbase + i * 4;
            float4 bw4 = *(const float4*)&sBW[d];

            float4 v = xrow[i];
            float4 w = make_float4(v.x * bw4.x, v.y * bw4.y, v.z * bw4.z, v.w * bw4.w);
            px += w.x * v.x + w.y * v.y + w.z * v.z + w.w * v.w;
            *(float4*)&sX[r * LDX + d] = w;              // store wx

            float4 u = yrow[i];
            py += u.x * u.x * bw4.x + u.y * u.y * bw4.y + u.z * u.z * bw4.z + u.w * u.w * bw4.w;
            *(float4*)&sY[r * LDX + d] = u;              // store raw y
        }
        sPartX[tid] = px;
        sPartY[tid] = py;
    }
    __syncthreads();
    if (tid < BM)        sX2[tid] = sPartX[2 * tid] + sPartX[2 * tid + 1];
    else                 sY2[tid - BM] = sPartY[2 * (tid - BM)] + sPartY[2 * (tid - BM) + 1];
    __syncthreads();

    // ---- per-wave tile: 64 rows x 32 cols (4x2 subtiles of 16x16) ----
    const int wave = tid >> 5;          // 0..7
    const int lane = tid & 31;
    const int m    = lane & 15;
    const int half = lane >> 4;
    const int wm   = wave >> 2;         // 0..1 -> row offset wm*64
    const int wn   = wave & 3;          // 0..3 -> col offset wn*32

    v8f acc[4][2] = {};                 // zero accumulators

    const float* pA = &sX[(wm * 64 + m) * LDX];
    const float* pB = &sY[(wn * 32 + m) * LDX];

    #pragma unroll
    for (int kk = 0; kk < 16; ++kk) {   // K = 64 in chunks of 4
        const int k0 = kk * 4 + half * 2;
        v2f a[4], b[2];
        #pragma unroll
        for (int i = 0; i < 4; ++i) a[i] = *(const v2f*)&pA[i * 16 * LDX + k0];
        #pragma unroll
        for (int j = 0; j < 2; ++j) b[j] = *(const v2f*)&pB[j * 16 * LDX + k0];
        #pragma unroll
        for (int i = 0; i < 4; ++i)
            #pragma unroll
            for (int j = 0; j < 2; ++j)
                acc[i][j] = __builtin_amdgcn_wmma_f32_16x16x4_f32(
                    false, a[i], false, b[j], (short)0, acc[i][j], false, false);
    }

    // ---- epilogue: pdist -> exp -> streaming store ----
    #pragma unroll
    for (int i = 0; i < 4; ++i) {
        #pragma unroll
        for (int j = 0; j < 2; ++j) {
            const int col_l = wn * 32 + j * 16 + m;
            const float y2v = sY2[col_l];
            const size_t col = (size_t)(blockCol + col_l);
            #pragma unroll
            for (int r = 0; r < 8; ++r) {
                const int row_l = wm * 64 + i * 16 + r + half * 8;
                const float c = acc[i][j][r];
                float pd = fmaf(-2.0f, c, sX2[row_l] + y2v);
                pd = fmaxf(pd, 0.0f);
                const float val = __expf(-0.5f * pd);
                __builtin_nontemporal_store(val,
                    &out[(size_t)(blockRow + row_l) * 8192 + col]);
            }
        }
    }
}

extern "C" void kernel_launch(void* const* d_in, const int* in_sizes, int n_in,
                              void* d_out, int out_size, void* d_ws, size_t ws_size,
                              hipStream_t stream) {
    const float* x   = (const float*)d_in[0];   // [8192*64]
    const float* y   = (const float*)d_in[1];   // [8192*64]
    const float* lbw = (const float*)d_in[2];   // [64]
    float* out = (float*)d_out;                 // [8192*8192]

    dim3 grid(8192 / BN, 8192 / BM);
    dim3 block(256);
    ARD_wmma_f32_kernel<<<grid, block, 0, stream>>>(x, y, lbw, out);
}